// ChamferDistance_22127671509748
// MI455X (gfx1250) — compile-verified
//
#include <hip/hip_runtime.h>

typedef __attribute__((ext_vector_type(2))) float v2f;
typedef __attribute__((ext_vector_type(8))) float v8f;

#define BB 4
#define DD 3
#define NN 4096
#define MM 4096
#define MSPLIT 4                        // m-range splits per (batch, n-block)
#define MT_PER_BLK (MM / 16 / MSPLIT)   // 64 m-tiles per wave sweep

// ---------------------------------------------------------------------------
// Kernel 1a: init row-min + col-min buffers (contiguous 2*B*4096 uints) to the
// +inf bit pattern. uint atomicMin on non-negative float bit patterns is exact
// and order-independent -> deterministic.
// ---------------------------------------------------------------------------
__global__ void cd_init(unsigned* __restrict__ mins, int n) {
    int i = blockIdx.x * blockDim.x + threadIdx.x;
    if (i < n) mins[i] = 0x7F800000u;  // +inf
}

// ---------------------------------------------------------------------------
// Kernel 1b: precompute xnorm[b][n] and the packed y tile ypack[b][m] =
// {y0, y1, y2, ||y||^2}. Packing lets the hot loop fetch everything it needs
// for one m-point with a single global_load_b128.
// ---------------------------------------------------------------------------
__global__ void cd_norms(const float* __restrict__ x, const float* __restrict__ y,
                         float* __restrict__ xnorm, float4* __restrict__ ypack) {
    int i = blockIdx.x * blockDim.x + threadIdx.x;
    if (i >= BB * NN) return;
    int b = i >> 12, idx = i & (NN - 1);
    const float* X = x + (size_t)b * DD * NN;
    float x0 = X[idx], x1 = X[NN + idx], x2 = X[2 * NN + idx];
    xnorm[i] = x0 * x0 + x1 * x1 + x2 * x2;
    const float* Y = y + (size_t)b * DD * MM;
    float y0 = Y[idx], y1 = Y[MM + idx], y2 = Y[2 * MM + idx];
    float4 w;
    w.x = y0; w.y = y1; w.z = y2;
    w.w = y0 * y0 + y1 * y1 + y2 * y2;
    ypack[i] = w;
}

// ---------------------------------------------------------------------------
// Kernel 2: main WMMA kernel.
// grid = (n-blocks=32, m-splits=4, batches=4). Each wave owns one 16-row
// n-tile and sweeps 64 m-tiles:  dist = ||x||^2 + ||y||^2 - 2*(x.y),
// cross term via V_WMMA_F32_16X16X4_F32 (K=4, d=3 zero-padded).
// Hot loop: 1x global_load_b128 + 2x v_cndmask + 1x v_wmma + packed epilogue
// + 1x global_atomic_min_u32. No DS ops, no exec divergence.
// ---------------------------------------------------------------------------
__global__ __launch_bounds__(256) void cd_main(const float* __restrict__ x,
                                               unsigned* __restrict__ rowmin,
                                               unsigned* __restrict__ colmin,
                                               const float* __restrict__ xnormg,
                                               const float4* __restrict__ ypack) {
    const int b    = blockIdx.z;
    const int wave = threadIdx.x >> 5;
    const int lane = threadIdx.x & 31;
    const int half = lane >> 4;       // 0: K=0,1 / rows 0..7 side; 1: K=2,3 / rows 8..15
    const int q    = lane & 15;
    const int n_base   = (blockIdx.x * 8 + wave) * 16;
    const int mt_begin = blockIdx.y * MT_PER_BLK;

    const float* X = x + (size_t)b * DD * NN;

    // A tile (16x4 f32, ISA layout): lanes 0-15 hold K=0/K=1 of row M=lane,
    // lanes 16-31 hold K=2/K=3 of row M=lane-16. K=3 is the d=3 zero pad.
    v2f a;
    {
        const int n_idx = n_base + q;
        float t0 = X[(half ? 2 * NN : 0) + n_idx];
        float t1 = X[NN + n_idx];
        a.x = t0;
        a.y = half ? 0.0f : t1;       // v_cndmask, no control flow
    }

    // Row norms for the 8 rows this half's C registers hold (M = r + 8*half).
    float xn[8];
    {
        const float* xrow = xnormg + ((size_t)b << 12) + n_base + half * 8;
#pragma unroll
        for (int r = 0; r < 8; ++r) xn[r] = xrow[r];
    }

    float rmin[8];
#pragma unroll
    for (int r = 0; r < 8; ++r) rmin[r] = 3.4e38f;

    const float4* Yp   = ypack + ((size_t)b << 12);
    unsigned*     colb = colmin + ((size_t)b << 12);

#pragma unroll 4
    for (int mt = mt_begin; mt < mt_begin + MT_PER_BLK; ++mt) {
        const int m_idx = (mt << 4) + q;
        // One b128 load delivers the B-tile operands and the column norm.
        float4 w = Yp[m_idx];
        v2f bm;
        bm.x = half ? w.z : w.x;      // v_cndmask
        bm.y = half ? 0.0f : w.y;     // v_cndmask (K=3 pad)
        float yn = w.w;               // ||y_m||^2

        v8f c = {0.f, 0.f, 0.f, 0.f, 0.f, 0.f, 0.f, 0.f};
        c = __builtin_amdgcn_wmma_f32_16x16x4_f32(
                /*neg_a=*/false, a, /*neg_b=*/false, bm,
                /*c_mod=*/(short)0, c, /*reuse_a=*/false, /*reuse_b=*/false);

        float cp = 3.4e38f;
#pragma unroll
        for (int r = 0; r < 8; ++r) {
            // element (M = r + 8*half, N = lane&15); clamp deferred (max(0,.)
            // commutes with min), saving 8 v_max per tile.
            float dch = fmaf(-2.0f, c[r], xn[r] + yn);
            rmin[r] = fminf(rmin[r], dch);  // running min over m for row M
            cp      = fminf(cp, dch);       // min over this half's 8 rows
        }
        cp = fmaxf(cp, 0.0f);
        // Both halves issue the atomic with their 8-row partial; the uint
        // atomic-min merges halves/waves/blocks in L2. No shuffle, no branch.
        atomicMin(&colb[m_idx], __float_as_uint(cp));
    }

    // Row mins: reduce across the 16 lanes of each half (the N columns).
#pragma unroll
    for (int mask = 1; mask < 16; mask <<= 1) {
#pragma unroll
        for (int r = 0; r < 8; ++r)
            rmin[r] = fminf(rmin[r], __shfl_xor(rmin[r], mask, 32));
    }
    // One lane per half merges its 8 finished rows (rows half*8 + 0..7) with
    // the other m-splits via atomic min (clamped here, once).
    if (q == 0) {
        unsigned* dst = rowmin + ((size_t)b << 12) + n_base + half * 8;
#pragma unroll
        for (int r = 0; r < 8; ++r)
            atomicMin(&dst[r], __float_as_uint(fmaxf(rmin[r], 0.0f)));
    }
}

// ---------------------------------------------------------------------------
// Kernel 3: deterministic reduction of row-mins + col-mins -> mean over batch.
// First 2*B*4096 words of ws are float bit patterns (all non-negative; every
// slot is written by construction).
// ---------------------------------------------------------------------------
__global__ __launch_bounds__(256) void cd_final(const float* __restrict__ ws,
                                                float* __restrict__ out) {
    __shared__ float s[256];
    float acc = 0.0f;
    const int total = BB * NN + BB * MM;
    for (int i = threadIdx.x; i < total; i += 256) acc += ws[i];
    s[threadIdx.x] = acc;
    __syncthreads();
    for (int k = 128; k > 0; k >>= 1) {
        if ((int)threadIdx.x < k) s[threadIdx.x] += s[threadIdx.x + k];
        __syncthreads();
    }
    if (threadIdx.x == 0) out[0] = s[0] * (1.0f / (float)BB);
}

// ---------------------------------------------------------------------------
extern "C" void kernel_launch(void* const* d_in, const int* in_sizes, int n_in,
                              void* d_out, int out_size, void* d_ws, size_t ws_size,
                              hipStream_t stream) {
    const float* x = (const float*)d_in[0];  // [B, D, N]
    const float* y = (const float*)d_in[1];  // [B, D, M]
    float* out = (float*)d_out;

    // Workspace layout (32-bit words, contiguous, 448 KB total):
    //   [0,      16384)  rowmin (uint bit patterns)
    //   [16384,  32768)  colmin (uint bit patterns)
    //   [32768,  49152)  xnorm  (float)
    //   [49152, 114688)  ypack  (float4 {y0,y1,y2,||y||^2}), 16B aligned
    unsigned* rowmin = (unsigned*)d_ws;
    unsigned* colmin = (unsigned*)d_ws + (size_t)BB * NN;
    float*    xnorm  = (float*)d_ws + 2 * (size_t)BB * NN;
    float4*   ypack  = (float4*)((float*)d_ws + 3 * (size_t)BB * NN);

    const int nmins = 2 * BB * NN;
    cd_init<<<(nmins + 255) / 256, 256, 0, stream>>>(rowmin, nmins);
    cd_norms<<<(BB * NN + 255) / 256, 256, 0, stream>>>(x, y, xnorm, ypack);

    // 8 waves per block, each wave owns one 16-row n-tile and one m-range:
    dim3 grid(NN / (16 * 8), MSPLIT, BB);
    cd_main<<<grid, 256, 0, stream>>>(x, rowmin, colmin, xnorm, ypack);

    cd_final<<<1, 256, 0, stream>>>((const float*)d_ws, out);
}